// AbstractEmbeddingSpace_26285199851866
// MI455X (gfx1250) — compile-verified
//
#include <hip/hip_runtime.h>

// Shapes from the reference
#define B_    8
#define D_    64
#define H_    128
#define W_    128
#define HW_   (H_ * W_)        // 16384
#define M_    1024
#define GROUP_ 32
#define EPS_  1e-15f

typedef __attribute__((ext_vector_type(2))) float v2f;
typedef __attribute__((ext_vector_type(4))) float v4f;
typedef __attribute__((ext_vector_type(8))) float v8f;

// bias[m] = <normals[m,:], offsets[m,:]>
__global__ __launch_bounds__(256) void bias_kernel(const float* __restrict__ normals,
                                                   const float* __restrict__ offsets,
                                                   float* __restrict__ bias) {
  int m = blockIdx.x * blockDim.x + threadIdx.x;
  if (m >= M_) return;
  float s = 0.f;
#pragma unroll 8
  for (int d = 0; d < D_; ++d)
    s += normals[m * D_ + d] * offsets[m * D_ + d];
  bias[m] = s;
}

// One block = 16 pixels of one batch image; all M=1024 rows.
// 8 waves, wave w owns M rows [w*128, w*128+128) as 8 WMMA 16x16 tiles.
__global__ __launch_bounds__(256, 1) void hsm_kernel(const float* __restrict__ x,
                                                     const float* __restrict__ normals,
                                                     const float* __restrict__ bias,
                                                     float* __restrict__ out) {
  __shared__ float xs[D_ * 16];   // staged x tile, layout [k][n]
  __shared__ float mbuf[8 * 16];  // per-wave per-column max

  const int tid  = threadIdx.x;
  const int wave = tid >> 5;
  const int lane = tid & 31;
  const int half = lane >> 4;     // 0: lanes 0-15, 1: lanes 16-31
  const int n    = lane & 15;     // column within tile

  const int b     = blockIdx.x >> 10;          // 1024 pixel-tiles per batch
  const int pbase = (blockIdx.x & 1023) << 4;  // 16 pixels
  const int mbase = wave << 7;                 // 128 rows per wave

  __builtin_prefetch(normals + (size_t)mbase * D_, 0, 0);

  // ---- stage X tile (D_ x 16) into LDS: xs[k*16 + n] = x[b, k, pbase+n] ----
#pragma unroll
  for (int i = 0; i < 4; ++i) {
    int idx = tid + 256 * i;
    int d = idx >> 4, c = idx & 15;
    xs[idx] = x[((size_t)(b * D_ + d)) * HW_ + pbase + c];
  }
  __syncthreads();

  // ---- GEMM: logits(128x16 per wave) = normals * X via f32 WMMA 16x16x4 ----
  v8f acc[8];
#pragma unroll
  for (int t = 0; t < 8; ++t) acc[t] = (v8f){};

#pragma unroll
  for (int kk = 0; kk < 16; ++kk) {
    const int kb = kk * 4 + half * 2;  // lanes 0-15: K=kb,kb+1 ; lanes 16-31: K=kb+2,kb+3
    v2f bfrag;
    bfrag.x = xs[kb * 16 + n];
    bfrag.y = xs[(kb + 1) * 16 + n];
#pragma unroll
    for (int t = 0; t < 8; ++t) {
      const int mA = mbase + t * 16 + n;  // A: lanes 0-15 -> M=lane, 16-31 -> M=lane-16
      v2f afrag = *(const v2f*)(normals + (size_t)mA * D_ + kb);
      acc[t] = __builtin_amdgcn_wmma_f32_16x16x4_f32(
          /*neg_a=*/false, afrag, /*neg_b=*/false, bfrag,
          /*c_mod=*/(short)0, acc[t], /*reuse_a=*/false, /*reuse_b=*/false);
    }
  }

  // ---- subtract bias (C/D layout: VGPR r holds M = r + 8*half within tile) ----
#pragma unroll
  for (int t = 0; t < 8; ++t) {
    const int base = mbase + t * 16 + half * 8;
    v4f b0 = *(const v4f*)(bias + base);
    v4f b1 = *(const v4f*)(bias + base + 4);
#pragma unroll
    for (int r = 0; r < 4; ++r) {
      acc[t][r]     -= b0[r];
      acc[t][r + 4] -= b1[r];
    }
  }

  // ---- global max over all M per column ----
  float mx = acc[0][0];
#pragma unroll
  for (int t = 0; t < 8; ++t)
#pragma unroll
    for (int r = 0; r < 8; ++r) mx = fmaxf(mx, acc[t][r]);
  mx = fmaxf(mx, __shfl_xor(mx, 16, 32));  // merge the two lane halves of column n
  if (lane < 16) mbuf[wave * 16 + n] = mx;
  __syncthreads();
  float gmx = mbuf[n];
#pragma unroll
  for (int w2 = 1; w2 < 8; ++w2) gmx = fmaxf(gmx, mbuf[w2 * 16 + n]);

  // ---- exp(logit - max) ----
#pragma unroll
  for (int t = 0; t < 8; ++t)
#pragma unroll
    for (int r = 0; r < 8; ++r) acc[t][r] = __expf(acc[t][r] - gmx);

  // ---- sibling-group (32 rows = 2 tiles) sums, never cross a wave ----
  float rz[4];
#pragma unroll
  for (int g = 0; g < 4; ++g) {
    float s = 0.f;
#pragma unroll
    for (int r = 0; r < 8; ++r) s += acc[2 * g][r] + acc[2 * g + 1][r];
    s += __shfl_xor(s, 16, 32);  // partner half holds rows 8..15 of each tile
    rz[g] = 1.0f / (s + EPS_);
  }

  // ---- write cond_probs: out[b, m, pbase+n] ----
#pragma unroll
  for (int t = 0; t < 8; ++t) {
    const float rc  = rz[t >> 1];
    const int mrow  = mbase + t * 16 + half * 8;
#pragma unroll
    for (int r = 0; r < 8; ++r) {
      const int m = mrow + r;
      out[((size_t)(b * M_ + m)) * HW_ + pbase + n] = acc[t][r] * rc;
    }
  }
}

extern "C" void kernel_launch(void* const* d_in, const int* in_sizes, int n_in,
                              void* d_out, int out_size, void* d_ws, size_t ws_size,
                              hipStream_t stream) {
  const float* x       = (const float*)d_in[0];  // (B,D,H,W) f32
  const float* normals = (const float*)d_in[1];  // (M,D) f32
  const float* offsets = (const float*)d_in[2];  // (M,D) f32
  // d_in[3] = sibmat (block-diagonal 32x32 ones; structure exploited directly)
  // d_in[4] = steps (unused by the reference math)
  float* out  = (float*)d_out;
  float* bias = (float*)d_ws;  // M_ floats of scratch

  bias_kernel<<<(M_ + 255) / 256, 256, 0, stream>>>(normals, offsets, bias);

  const int grid = B_ * (HW_ / 16);  // 8192 blocks
  hsm_kernel<<<grid, 256, 0, stream>>>(x, normals, bias, out);
}